// NeuralAutomaton_26912265077324
// MI455X (gfx1250) — compile-verified
//
#include <hip/hip_runtime.h>

typedef __attribute__((ext_vector_type(2))) float v2f;
typedef __attribute__((ext_vector_type(8))) float v8f;

#define NSTATE 10
#define NSYM   10
#define NCLS   2
#define SEQ    8192
#define BATCH  256
#define CHUNKS 32
#define CHUNK_LEN (SEQ / CHUNKS)   // 256

// K is truncated to 12 (3 chunks of 4): columns/rows >= 10 of the padded
// 16x16 operands are structurally zero, so the 4th K-chunk contributes nothing.
#define NKC   3
#define NREG  (2 * NKC)            // 6 A/B regs per lane

// C += A(16xK) * B(Kx16) as NKC chained V_WMMA_F32_16X16X4_F32.
// a[NREG]: A-layout regs (per lane: A(lo, kidx(j)))
// b[NREG]: B-layout regs (per lane: B(kidx(j), lo))
__device__ __forceinline__ v8f wmma_16x16_f32(const float a[NREG], const float b[NREG], v8f c) {
#pragma unroll
  for (int kc = 0; kc < NKC; ++kc) {
    v2f av; av.x = a[2 * kc]; av.y = a[2 * kc + 1];
    v2f bv; bv.x = b[2 * kc]; bv.y = b[2 * kc + 1];
    c = __builtin_amdgcn_wmma_f32_16x16x4_f32(false, av, false, bv, (short)0, c,
                                              false, false);
  }
  return c;
}

// Kernel 1: each wave (one 32-thread workgroup) computes the ordered product
// G = M_{t0} * M_{t0+1} * ... * M_{t0+CHUNK_LEN-1} for one (batch, chunk),
// where M_t = sum_s x[b,t,s] * P[s], zero-padded 10x10 -> 16x16.
__global__ void __launch_bounds__(32)
automaton_chunk_kernel(const float* __restrict__ x, const float* __restrict__ P,
                       float* __restrict__ gprod) {
  __shared__ float tile[256];
  const int lane = threadIdx.x;
  const int lo = lane & 15;
  const int hi = lane >> 4;
  const int b = blockIdx.x / CHUNKS;
  const int chunk = blockIdx.x % CHUNKS;

  int kidx[NREG];
#pragma unroll
  for (int j = 0; j < NREG; ++j) kidx[j] = 4 * (j >> 1) + 2 * hi + (j & 1);

  // P resident in VGPRs in B-layout, zero-padded outside the 10x10 block.
  float pb[NSYM][NREG];
#pragma unroll
  for (int s = 0; s < NSYM; ++s)
#pragma unroll
    for (int j = 0; j < NREG; ++j)
      pb[s][j] = (kidx[j] < NSTATE && lo < NSTATE)
                     ? P[s * NSTATE * NSTATE + kidx[j] * NSTATE + lo]
                     : 0.0f;

  // acc = identity (top-left 12x12 portion suffices), in A-layout.
  float acc[NREG];
#pragma unroll
  for (int j = 0; j < NREG; ++j) acc[j] = (lo == kidx[j]) ? 1.0f : 0.0f;

  const float* xp = x + (size_t)(b * SEQ + chunk * CHUNK_LEN) * NSYM;

  for (int t = 0; t < CHUNK_LEN; ++t) {
    // Build M_t in B-layout: m[j] = sum_s x_s * P[s][kidx(j)][lo].
    float m[NREG];
#pragma unroll
    for (int j = 0; j < NREG; ++j) m[j] = 0.0f;
#pragma unroll
    for (int s = 0; s < NSYM; ++s) {
      const float xs = xp[t * NSYM + s];  // wave-uniform -> scalar load
#pragma unroll
      for (int j = 0; j < NREG; ++j) m[j] = __builtin_fmaf(xs, pb[s][j], m[j]);
    }

    v8f d = {0.f, 0.f, 0.f, 0.f, 0.f, 0.f, 0.f, 0.f};
    d = wmma_16x16_f32(acc, m, d);  // d = acc @ M_t

    // Relayout D -> A through LDS for the next chain step.
    __syncthreads();
#pragma unroll
    for (int v = 0; v < 8; ++v) tile[(v + 8 * hi) * 16 + lo] = d[v];
    __syncthreads();
#pragma unroll
    for (int j = 0; j < NREG; ++j) acc[j] = tile[lo * 16 + kidx[j]];
    __syncthreads();
  }

  // Store chunk product row-major 16x16 (rows 0..15, cols 0..11 written;
  // cols >= 12 feed only discarded output columns downstream).
  float* gp = gprod + (size_t)(b * CHUNKS + chunk) * 256;
#pragma unroll
  for (int j = 0; j < NREG; ++j) gp[lo * 16 + kidx[j]] = acc[j];
}

// Kernel 2: one wave per batch chains the 32 chunk products in order, then
// out[b,c] = sum_{n,m} start[n] * prod[n][m] * accept[m][c].
__global__ void __launch_bounds__(32)
automaton_reduce_kernel(const float* __restrict__ gprod,
                        const float* __restrict__ start,
                        const float* __restrict__ accept,
                        float* __restrict__ out) {
  __shared__ float tile[256];
  const int lane = threadIdx.x;
  const int lo = lane & 15;
  const int hi = lane >> 4;
  const int b = blockIdx.x;

  int kidx[NREG];
#pragma unroll
  for (int j = 0; j < NREG; ++j) kidx[j] = 4 * (j >> 1) + 2 * hi + (j & 1);

  const float* gp = gprod + (size_t)b * CHUNKS * 256;

  // acc = G_0 loaded in A-layout (cols <= 11 only; rest structurally zero).
  float acc[NREG];
#pragma unroll
  for (int j = 0; j < NREG; ++j) acc[j] = gp[lo * 16 + kidx[j]];

  for (int cI = 1; cI < CHUNKS; ++cI) {
    const float* g = gp + (size_t)cI * 256;
    float m[NREG];
#pragma unroll
    for (int j = 0; j < NREG; ++j) m[j] = g[kidx[j] * 16 + lo];  // B-layout

    v8f d = {0.f, 0.f, 0.f, 0.f, 0.f, 0.f, 0.f, 0.f};
    d = wmma_16x16_f32(acc, m, d);

    __syncthreads();
#pragma unroll
    for (int v = 0; v < 8; ++v) tile[(v + 8 * hi) * 16 + lo] = d[v];
    __syncthreads();
#pragma unroll
    for (int j = 0; j < NREG; ++j) acc[j] = tile[lo * 16 + kidx[j]];
    __syncthreads();
  }

  // Publish final product to LDS row-major, then 2 lanes do the tiny bilinear.
#pragma unroll
  for (int j = 0; j < NREG; ++j) tile[lo * 16 + kidx[j]] = acc[j];
  __syncthreads();

  if (lane < NCLS) {
    float r = 0.0f;
#pragma unroll
    for (int n = 0; n < NSTATE; ++n) {
      const float sv = start[n];
#pragma unroll
      for (int mm = 0; mm < NSTATE; ++mm)
        r = __builtin_fmaf(sv * tile[n * 16 + mm], accept[mm * NCLS + lane], r);
    }
    out[b * NCLS + lane] = r;
  }
}

extern "C" void kernel_launch(void* const* d_in, const int* in_sizes, int n_in,
                              void* d_out, int out_size, void* d_ws, size_t ws_size,
                              hipStream_t stream) {
  (void)in_sizes; (void)n_in; (void)out_size; (void)ws_size;
  const float* x      = (const float*)d_in[0];  // [256, 8192, 10]
  const float* P      = (const float*)d_in[1];  // [10, 10, 10]
  const float* start  = (const float*)d_in[2];  // [10]
  const float* accept = (const float*)d_in[3];  // [10, 2]
  float* out = (float*)d_out;                   // [256, 2]
  float* gprod = (float*)d_ws;                  // 256*32*256 floats = 8 MB

  automaton_chunk_kernel<<<BATCH * CHUNKS, 32, 0, stream>>>(x, P, gprod);
  automaton_reduce_kernel<<<BATCH, 32, 0, stream>>>(gprod, start, accept, out);
}